// NodeFeatures_81484119539776
// MI455X (gfx1250) — compile-verified
//
#include <hip/hip_runtime.h>
#include <hip/hip_bf16.h>

// ---------------------------------------------------------------------------
// MI455X (gfx1250, wave32) implementation.
//
// Traffic analysis: edge stream (2 x 205MB edge_features + 2 x 12.8MB index)
// dominates; node arrays (h1,h2,denom,agg = 51MB) stay L2-resident (192MB L2)
// so edge streams use non-temporal loads. The two tiny MLPs (6.6 GFLOP) run
// on the f32 WMMA path (V_WMMA_F32_16X16X4_F32) for exact-precision matmul.
// ---------------------------------------------------------------------------

typedef float v2f __attribute__((ext_vector_type(2)));
typedef float v8f __attribute__((ext_vector_type(8)));

#define FD 32            // feature dim (c_in1 == c_out2)
#define FH 128           // hidden dim  (c_out1)
#define YS (FH + 4)      // padded LDS stride (breaks bank conflicts)

__device__ __forceinline__ float sigmoidf_fast(float v) {
    return 1.0f / (1.0f + __expf(-v));
}

// ---------------------------------------------------------------------------
// Kernel 1: h1 = FCNN_one(x), h2 = FCNN_two(x) via f32 WMMA.
// One wave per 16-node tile, 4 waves / block.
//
// WMMA f32 16x16x4 fragment layouts (wave32):
//   A (16x4):  lanes 0-15 = rows, v0/v1 = K0/K1; lanes 16-31 = K2/K3
//   B (4x16):  lanes 0-15 = cols, v0/v1 = K0/K1; lanes 16-31 = K2/K3
//   C/D (16x16): lane%16 = col; vgpr m = row m (lanes<16) / row m+8 (lanes>=16)
// ---------------------------------------------------------------------------
__global__ __launch_bounds__(128) void fcnn_wmma_kernel(
    const float* __restrict__ x,
    const float* __restrict__ W1a, const float* __restrict__ b1a,
    const float* __restrict__ W2a, const float* __restrict__ b2a,
    const float* __restrict__ W1b, const float* __restrict__ b1b,
    const float* __restrict__ W2b, const float* __restrict__ b2b,
    float* __restrict__ h1, float* __restrict__ h2, int N)
{
    __shared__ float Ystage[4][16 * YS];     // per-wave Y staging (relu(xW1+b1))
    const int lane  = threadIdx.x & 31;
    const int wave  = threadIdx.x >> 5;
    const int r     = lane & 15;             // row (A) or col (B/C) index
    const int khalf = lane >> 4;             // which K-pair this half-wave owns
    const int tile  = blockIdx.x * 4 + wave;
    const int row   = tile * 16 + r;
    const int lrow  = row < N ? row : (N - 1);   // clamp loads, guard stores

    // A fragments for the 16x32 X tile: 8 K-steps of 4.
    v2f a[8];
#pragma unroll
    for (int kk = 0; kk < 8; ++kk) {
        const float* p = x + (size_t)lrow * FD + kk * 4 + khalf * 2;
        a[kk].x = p[0];
        a[kk].y = p[1];
    }
    float* Y = Ystage[wave];

#pragma unroll
    for (int pass = 0; pass < 2; ++pass) {
        const float* W1 = pass ? W1b : W1a;
        const float* B1 = pass ? b1b : b1a;
        const float* W2 = pass ? W2b : W2a;
        const float* B2 = pass ? b2b : b2a;
        float* hout     = pass ? h2  : h1;

        // ---- Layer 1: [16xFD] @ [FDxFH] + b, ReLU -> Y in LDS ----
#pragma unroll
        for (int nt = 0; nt < 8; ++nt) {
            const int col = nt * 16 + r;
            const float bias = B1[col];
            v8f c;
#pragma unroll
            for (int m = 0; m < 8; ++m) c[m] = bias;
#pragma unroll
            for (int kk = 0; kk < 8; ++kk) {
                const int k0 = kk * 4 + khalf * 2;
                v2f b;
                b.x = W1[k0 * FH + col];
                b.y = W1[(k0 + 1) * FH + col];
                c = __builtin_amdgcn_wmma_f32_16x16x4_f32(
                        false, a[kk], false, b, (short)0, c, false, false);
            }
#pragma unroll
            for (int m = 0; m < 8; ++m) {
                float v = c[m] > 0.0f ? c[m] : 0.0f;
                Y[(m + khalf * 8) * YS + col] = v;   // per-wave region: no barrier
            }
        }

        // ---- Layer 2: [16xFH] @ [FHxFD] + b -> hout ----
#pragma unroll
        for (int nt = 0; nt < 2; ++nt) {
            const int col = nt * 16 + r;
            const float bias = B2[col];
            v8f c;
#pragma unroll
            for (int m = 0; m < 8; ++m) c[m] = bias;
#pragma unroll 8
            for (int kk = 0; kk < 32; ++kk) {
                const int k0 = kk * 4 + khalf * 2;
                v2f av, b;
                av.x = Y[r * YS + k0];
                av.y = Y[r * YS + k0 + 1];
                b.x = W2[k0 * FD + col];
                b.y = W2[(k0 + 1) * FD + col];
                c = __builtin_amdgcn_wmma_f32_16x16x4_f32(
                        false, av, false, b, (short)0, c, false, false);
            }
#pragma unroll
            for (int m = 0; m < 8; ++m) {
                const int orow = tile * 16 + m + khalf * 8;
                if (orow < N) hout[(size_t)orow * FD + col] = c[m];
            }
        }
    }
}

// ---------------------------------------------------------------------------
// Kernel 2: denom <- EPS_DENOM, agg <- 0   (ws is poisoned before timing)
// ---------------------------------------------------------------------------
__global__ __launch_bounds__(256) void init_kernel(
    float* __restrict__ denom, float* __restrict__ agg, int n)
{
    int t = blockIdx.x * blockDim.x + threadIdx.x;
    if (t < n) { denom[t] = 1e-7f; agg[t] = 0.0f; }
}

// ---------------------------------------------------------------------------
// Kernel 3: denom[i] += sigmoid(ef[e]) for both endpoints of every edge.
// Thread = (edge, feature). Wave handles one edge -> coalesced 128B stream,
// 32 consecutive f32 atomics per endpoint.
// ---------------------------------------------------------------------------
__global__ __launch_bounds__(256) void edge_denom_kernel(
    const float* __restrict__ ef, const int* __restrict__ eidx,
    float* __restrict__ denom, int E)
{
    int t = blockIdx.x * blockDim.x + threadIdx.x;
    if (t >= E * FD) return;
    const int e = t >> 5, d = t & 31;
    const float s = sigmoidf_fast(__builtin_nontemporal_load(ef + t));
    const int src = __builtin_nontemporal_load(eidx + e);
    const int dst = __builtin_nontemporal_load(eidx + E + e);
    atomicAdd(&denom[src * FD + d], s);
    atomicAdd(&denom[dst * FD + d], s);
}

// ---------------------------------------------------------------------------
// Kernel 4: agg[seg] += (s / denom[seg]) * h2[nbr], both directions.
// Recomputes sigmoid (cheaper than round-tripping 205MB of s through HBM).
// ---------------------------------------------------------------------------
__global__ __launch_bounds__(256) void edge_msg_kernel(
    const float* __restrict__ ef, const int* __restrict__ eidx,
    const float* __restrict__ denom, const float* __restrict__ h2,
    float* __restrict__ agg, int E)
{
    int t = blockIdx.x * blockDim.x + threadIdx.x;
    if (t >= E * FD) return;
    const int e = t >> 5, d = t & 31;
    const float s = sigmoidf_fast(__builtin_nontemporal_load(ef + t));
    const int src = __builtin_nontemporal_load(eidx + e);
    const int dst = __builtin_nontemporal_load(eidx + E + e);
    const int is = src * FD + d;
    const int id = dst * FD + d;
    atomicAdd(&agg[is], (s / denom[is]) * h2[id]);
    atomicAdd(&agg[id], (s / denom[id]) * h2[is]);
}

// ---------------------------------------------------------------------------
// Kernel 5: h = h1 + agg; InstanceNorm over D=32 (biased var); out = x+relu.
// One wave per node; wave32 butterfly reduction via __shfl_xor.
// ---------------------------------------------------------------------------
__global__ __launch_bounds__(256) void norm_kernel(
    const float* __restrict__ x, const float* __restrict__ h1,
    const float* __restrict__ agg, float* __restrict__ out, int N)
{
    int t = blockIdx.x * blockDim.x + threadIdx.x;
    const int n = t >> 5;
    if (n >= N) return;
    const float h = h1[t] + agg[t];
    float sum = h, sq = h * h;
#pragma unroll
    for (int off = 16; off >= 1; off >>= 1) {
        sum += __shfl_xor(sum, off, 32);
        sq  += __shfl_xor(sq,  off, 32);
    }
    const float mu  = sum * (1.0f / FD);
    const float var = sq * (1.0f / FD) - mu * mu;
    const float hn  = (h - mu) * rsqrtf(var + 1e-5f);
    out[t] = x[t] + (hn > 0.0f ? hn : 0.0f);
}

// ---------------------------------------------------------------------------
extern "C" void kernel_launch(void* const* d_in, const int* in_sizes, int n_in,
                              void* d_out, int out_size, void* d_ws, size_t ws_size,
                              hipStream_t stream) {
    const float* x   = (const float*)d_in[0];
    const float* ef  = (const float*)d_in[1];
    const float* W1a = (const float*)d_in[2];
    const float* b1a = (const float*)d_in[3];
    const float* W2a = (const float*)d_in[4];
    const float* b2a = (const float*)d_in[5];
    const float* W1b = (const float*)d_in[6];
    const float* b1b = (const float*)d_in[7];
    const float* W2b = (const float*)d_in[8];
    const float* b2b = (const float*)d_in[9];
    const int*  eidx = (const int*)d_in[10];      // edge_index, [2,E] int32
    float* out = (float*)d_out;

    const int N = in_sizes[0] / FD;
    const int E = in_sizes[10] / 2;

    float* h1    = (float*)d_ws;                  // [N, FD]
    float* h2    = h1    + (size_t)N * FD;        // [N, FD]
    float* denom = h2    + (size_t)N * FD;        // [N, FD]
    float* agg   = denom + (size_t)N * FD;        // [N, FD]

    const int tiles = (N + 15) / 16;
    fcnn_wmma_kernel<<<(tiles + 3) / 4, 128, 0, stream>>>(
        x, W1a, b1a, W2a, b2a, W1b, b1b, W2b, b2b, h1, h2, N);

    init_kernel<<<(N * FD + 255) / 256, 256, 0, stream>>>(denom, agg, N * FD);

    const int etot = E * FD;
    edge_denom_kernel<<<(etot + 255) / 256, 256, 0, stream>>>(ef, eidx, denom, E);
    edge_msg_kernel<<<(etot + 255) / 256, 256, 0, stream>>>(ef, eidx, denom, h2, agg, E);

    norm_kernel<<<(N * FD + 255) / 256, 256, 0, stream>>>(x, h1, agg, out, N);
}